// GCN_10058813407376
// MI455X (gfx1250) — compile-verified
//
#include <hip/hip_runtime.h>

typedef __attribute__((ext_vector_type(2))) float v2f;
typedef __attribute__((ext_vector_type(8))) float v8f;

// ---------------------------------------------------------------------------
// Degree / normalization:  deg[i] = 1 (self loop) + #edges with dst==i
// then dinv[i] = rsqrt(deg[i])
// ---------------------------------------------------------------------------
__global__ void k_init_deg(float* deg, int N) {
  int i = blockIdx.x * blockDim.x + threadIdx.x;
  if (i < N) deg[i] = 1.0f;
}

__global__ void k_count_deg(const int* __restrict__ dst, float* deg, int E) {
  int e = blockIdx.x * blockDim.x + threadIdx.x;
  if (e < E) atomicAdd(&deg[dst[e]], 1.0f);
}

__global__ void k_rsqrt_inplace(float* deg, int N) {
  int i = blockIdx.x * blockDim.x + threadIdx.x;
  if (i < N) deg[i] = rsqrtf(deg[i]);
}

// ---------------------------------------------------------------------------
// Zero-pad W [K x Nin] -> Wp [K x Npad]  (so GEMM B-loads are unconditional)
// ---------------------------------------------------------------------------
__global__ void k_pad_w(const float* __restrict__ W, float* __restrict__ Wp,
                        int K, int Nin, int Npad) {
  int i = blockIdx.x * blockDim.x + threadIdx.x;
  if (i >= K * Npad) return;
  int k = i / Npad, n = i - k * Npad;
  Wp[i] = (n < Nin) ? W[k * Nin + n] : 0.f;
}

// ---------------------------------------------------------------------------
// GEMM: Y[N x YN] = (relu? max(X,0):X)[N x K] @ W[K x WN]   (first YN cols)
// One wave per 16x16 tile, V_WMMA_F32_16X16X4_F32, K stepped by 4.
// WN is a multiple of 16 (W padded) -> NO conditional loads in the K-loop.
// Only the final store masks col < YN (outside the loop).
// A layout (16x4 f32): lane L -> M=L%16 ; VGPR0=K {0|2}, VGPR1=K {1|3}
// B layout (4x16 f32): lane L -> N=L%16 ; VGPR0=K {0|2}, VGPR1=K {1|3}
// C/D layout: VGPR r, lanes 0-15 -> M=r, lanes 16-31 -> M=r+8 ; N = L%16
// ---------------------------------------------------------------------------
__global__ void k_gemm_wmma(const float* __restrict__ X, const float* __restrict__ W,
                            float* __restrict__ Y, int N, int K, int WN, int YN,
                            int colTiles, int relu) {
  int waveId = (int)((blockIdx.x * blockDim.x + threadIdx.x) >> 5);
  int lane   = threadIdx.x & 31;
  int totalTiles = (N >> 4) * colTiles;
  if (waveId >= totalTiles) return;               // wave-uniform exit

  int tileRow = waveId / colTiles;
  int tileCol = waveId - tileRow * colTiles;
  int m     = (tileRow << 4) + (lane & 15);       // A row for this lane
  int col   = (tileCol << 4) + (lane & 15);       // B / D column for this lane
  int khalf = (lane >> 4) << 1;                   // 0 (lanes 0-15) or 2 (16-31)

  const float* __restrict__ ap = X + (size_t)m * K + khalf;       // A pair ptr
  const float* __restrict__ wp = W + (size_t)khalf * WN + col;    // B ptr
  const size_t wstep = (size_t)4 * WN;

  v8f c = {0.f, 0.f, 0.f, 0.f, 0.f, 0.f, 0.f, 0.f};

  if (relu) {
    #pragma unroll 4
    for (int k = 0; k < K; k += 4) {
      v2f a = *(const v2f*)ap;
      a.x = fmaxf(a.x, 0.f); a.y = fmaxf(a.y, 0.f);
      v2f b; b.x = wp[0]; b.y = wp[WN];
      c = __builtin_amdgcn_wmma_f32_16x16x4_f32(false, a, false, b,
                                                (short)0, c, false, false);
      ap += 4; wp += wstep;
    }
  } else {
    #pragma unroll 4
    for (int k = 0; k < K; k += 4) {
      v2f a = *(const v2f*)ap;
      v2f b; b.x = wp[0]; b.y = wp[WN];
      c = __builtin_amdgcn_wmma_f32_16x16x4_f32(false, a, false, b,
                                                (short)0, c, false, false);
      ap += 4; wp += wstep;
    }
  }

  if (col < YN) {                                  // mask only for YN=40 tile 2
    int rbase = (tileRow << 4) + ((lane < 16) ? 0 : 8);
    float* __restrict__ yp = Y + (size_t)rbase * YN + col;
    #pragma unroll
    for (int r = 0; r < 8; ++r)
      yp[(size_t)r * YN] = c[r];
  }
}

// ---------------------------------------------------------------------------
// out[i,f] = dinv[i]^2 * h[i,f] + bias[f]  (self-loop term + bias; fully
// initializes the aggregation buffer)
// ---------------------------------------------------------------------------
__global__ void k_self_init(const float* __restrict__ h, const float* __restrict__ dinv,
                            const float* __restrict__ bias, float* __restrict__ out,
                            int N, int F) {
  long long i = (long long)blockIdx.x * blockDim.x + threadIdx.x;
  long long total = (long long)N * F;
  if (i >= total) return;
  int node = (int)(i / F);
  int f    = (int)(i - (long long)node * F);
  float di = dinv[node];
  out[i] = di * di * h[i] + bias[f];
}

// ---------------------------------------------------------------------------
// One wave per edge: out[dst] += dinv[src]*dinv[dst] * h[src]  (F even, <=64)
// Lane handles adjacent channel pair {2*lane, 2*lane+1}: one b64 load,
// two global_atomic_add_f32. Rows are L2-resident.
// ---------------------------------------------------------------------------
__global__ void k_edge_agg(const float* __restrict__ h, const int* __restrict__ src,
                           const int* __restrict__ dst, const float* __restrict__ dinv,
                           float* __restrict__ out, int E, int F) {
  int e    = (int)((blockIdx.x * blockDim.x + threadIdx.x) >> 5);
  int lane = threadIdx.x & 31;
  if (e >= E) return;
  int s = src[e], d = dst[e];
  float coef = dinv[s] * dinv[d];
  int c0 = lane << 1;
  if (c0 < F) {
    v2f v = *(const v2f*)(h + (size_t)s * F + c0);
    float* __restrict__ od = out + (size_t)d * F + c0;
    atomicAdd(&od[0], coef * v.x);
    atomicAdd(&od[1], coef * v.y);
  }
}

// ---------------------------------------------------------------------------
// log_softmax per node row (F=40), one wave per node, wave32 shuffle reduce.
// ---------------------------------------------------------------------------
__global__ void k_log_softmax(const float* __restrict__ in, float* __restrict__ out,
                              int N, int F) {
  int node = (int)((blockIdx.x * blockDim.x + threadIdx.x) >> 5);
  int lane = threadIdx.x & 31;
  if (node >= N) return;
  const float* __restrict__ row = in + (size_t)node * F;
  float v0 = (lane < F)      ? row[lane]      : -__builtin_inff();
  float v1 = (lane + 32 < F) ? row[lane + 32] : -__builtin_inff();
  float m = fmaxf(v0, v1);
  #pragma unroll
  for (int off = 16; off; off >>= 1) m = fmaxf(m, __shfl_xor(m, off, 32));
  float s = 0.f;
  if (lane < F)      s += expf(v0 - m);
  if (lane + 32 < F) s += expf(v1 - m);
  #pragma unroll
  for (int off = 16; off; off >>= 1) s += __shfl_xor(s, off, 32);
  float lse = m + logf(s);
  float* __restrict__ orow = out + (size_t)node * F;
  if (lane < F)      orow[lane]      = v0 - lse;
  if (lane + 32 < F) orow[lane + 32] = v1 - lse;
}

// ---------------------------------------------------------------------------
extern "C" void kernel_launch(void* const* d_in, const int* in_sizes, int n_in,
                              void* d_out, int out_size, void* d_ws, size_t ws_size,
                              hipStream_t stream) {
  const float* x  = (const float*)d_in[0];
  const int*   ei = (const int*)d_in[1];   // [2, E] int32
  const float* W1 = (const float*)d_in[2];
  const float* b1 = (const float*)d_in[3];
  const float* W2 = (const float*)d_in[4];
  const float* b2 = (const float*)d_in[5];
  const float* W3 = (const float*)d_in[6];
  const float* b3 = (const float*)d_in[7];

  const int IN = 128, HID = 64, OUT = 40, OUTP = 48;
  const int N = in_sizes[0] / IN;          // 100000 (multiple of 16)
  const int E = in_sizes[1] / 2;           // 1250000
  const int* src = ei;
  const int* dst = ei + E;

  float* ws    = (float*)d_ws;
  float* dinv  = ws;                              // N
  float* bufA  = dinv + N;                        // N*64
  float* bufB  = bufA + (size_t)N * HID;          // N*64
  float* buf40 = bufB + (size_t)N * HID;          // N*40
  float* W3pad = buf40 + (size_t)N * OUT;         // 64*48
  float* outp  = (float*)d_out;                   // N*40

  const int T = 256;                              // 8 waves / block (wave32)
  const int wavesPerBlock = T / 32;

  // normalization coefficients
  k_init_deg<<<(N + T - 1) / T, T, 0, stream>>>(dinv, N);
  k_count_deg<<<(E + T - 1) / T, T, 0, stream>>>(dst, dinv, E);
  k_rsqrt_inplace<<<(N + T - 1) / T, T, 0, stream>>>(dinv, N);

  // pad W3 (64x40 -> 64x48, zero fill)
  k_pad_w<<<(HID * OUTP + T - 1) / T, T, 0, stream>>>(W3, W3pad, HID, OUT, OUTP);

  auto gemm = [&](const float* X, const float* W, float* Y, int K, int WN, int YN,
                  int relu) {
    int colTiles = (YN + 15) / 16;
    int tiles = (N >> 4) * colTiles;
    int blocks = (tiles + wavesPerBlock - 1) / wavesPerBlock;
    k_gemm_wmma<<<blocks, T, 0, stream>>>(X, W, Y, N, K, WN, YN, colTiles, relu);
  };
  auto selfinit = [&](const float* h, const float* b, float* o, int F) {
    long long total = (long long)N * F;
    k_self_init<<<(int)((total + T - 1) / T), T, 0, stream>>>(h, dinv, b, o, N, F);
  };
  auto edgeagg = [&](const float* h, float* o, int F) {
    long long threads = (long long)E * 32;
    k_edge_agg<<<(int)((threads + T - 1) / T), T, 0, stream>>>(h, src, dst, dinv, o, E, F);
  };

  // Layer 1: h1 = x @ W1 ; a1 = Agg(h1) + b1
  gemm(x, W1, bufA, IN, HID, HID, /*relu=*/0);
  selfinit(bufA, b1, bufB, HID);
  edgeagg(bufA, bufB, HID);

  // Layer 2: h2 = relu(a1) @ W2 ; a2 = Agg(h2) + b2
  gemm(bufB, W2, bufA, HID, HID, HID, /*relu=*/1);
  selfinit(bufA, b2, bufB, HID);
  edgeagg(bufA, bufB, HID);

  // Layer 3: h3 = relu(a2) @ W3 ; a3 = Agg(h3) + b3  (padded W, masked store)
  gemm(bufB, W3pad, bufA, HID, OUTP, OUT, /*relu=*/1);
  selfinit(bufA, b3, buf40, OUT);
  edgeagg(bufA, buf40, OUT);

  // log_softmax -> d_out
  {
    long long threads = (long long)N * 32;
    k_log_softmax<<<(int)((threads + T - 1) / T), T, 0, stream>>>(buf40, outp, N, OUT);
  }
}